// AttentionSAGEConv_64295660421655
// MI455X (gfx1250) — compile-verified
//
#include <hip/hip_runtime.h>

typedef float v2f __attribute__((ext_vector_type(2)));
typedef float v8f __attribute__((ext_vector_type(8)));

#define LEAKY 0.2f
#define SCALE 0.17677669529663687f  // 32^-0.5

// ---- monotone float<->uint mapping so we can use integer atomicMax for a
// ---- global float max (works for negatives; init value 0u < encode(-inf)).
__device__ __forceinline__ unsigned fenc(float f) {
  unsigned u = __float_as_uint(f);
  return (u & 0x80000000u) ? ~u : (u | 0x80000000u);
}
__device__ __forceinline__ float fdec(unsigned u) {
  return __uint_as_float((u & 0x80000000u) ? (u & 0x7FFFFFFFu) : ~u);
}

// ---------------------------------------------------------------------------
// Generic fp32 WMMA GEMM: C[nrows,ncols] = A[nrows,kdim] @ W[kdim,ncols]
//                                        (+ A2 @ W2) (+ bias) (opt. relu)
// One wave per 16x16 output tile; K-loop of v_wmma_f32_16x16x4_f32.
// A-matrix VGPR layout (16x4 f32): lanes 0-15 hold M=lane, {K=k,k+1};
//                                  lanes 16-31 hold M=lane-16, {K=k+2,k+3}.
// B (4x16) mirrors it; C/D: VGPR r -> row r (lanes 0-15) / row r+8 (16-31).
// ---------------------------------------------------------------------------
__global__ void wmma_gemm_f32(const float* __restrict__ A,
                              const float* __restrict__ W,
                              const float* __restrict__ A2,
                              const float* __restrict__ W2,
                              const float* __restrict__ bias,
                              float* __restrict__ C,
                              int nrows, int kdim, int ncols, int do_relu) {
  const int lane  = threadIdx.x & 31;
  const int wave  = blockIdx.x * (blockDim.x >> 5) + (threadIdx.x >> 5);
  const int ctile = ncols >> 4;
  const int rt = wave / ctile;
  const int ct = wave % ctile;
  if (rt * 16 >= nrows) return;              // wave-uniform: EXEC stays all-1s
  const int row0 = rt << 4, col0 = ct << 4;
  const int half = lane >> 4;                // 0: K=k,k+1   1: K=k+2,k+3
  const int l    = lane & 15;

  v8f c = {0.f, 0.f, 0.f, 0.f, 0.f, 0.f, 0.f, 0.f};

  {
    const float* ap = A + (size_t)(row0 + l) * kdim + 2 * half;
    const float* bp = W + (size_t)(2 * half) * ncols + col0 + l;
#pragma unroll 8
    for (int k = 0; k < kdim; k += 4) {
      v2f a = *(const v2f*)ap;               // {K+0,K+1} or {K+2,K+3}, 8B aligned
      v2f b; b.x = bp[0]; b.y = bp[ncols];
      c = __builtin_amdgcn_wmma_f32_16x16x4_f32(false, a, false, b,
                                                (short)0, c, false, false);
      ap += 4; bp += 4 * (size_t)ncols;
    }
  }
  if (A2 != nullptr) {                       // fused concat([x,agg2]) @ Wm
    const float* ap = A2 + (size_t)(row0 + l) * kdim + 2 * half;
    const float* bp = W2 + (size_t)(2 * half) * ncols + col0 + l;
#pragma unroll 8
    for (int k = 0; k < kdim; k += 4) {
      v2f a = *(const v2f*)ap;
      v2f b; b.x = bp[0]; b.y = bp[ncols];
      c = __builtin_amdgcn_wmma_f32_16x16x4_f32(false, a, false, b,
                                                (short)0, c, false, false);
      ap += 4; bp += 4 * (size_t)ncols;
    }
  }
  if (bias != nullptr) {
    float bv = bias[col0 + l];
#pragma unroll
    for (int r = 0; r < 8; ++r) c[r] += bv;
  }
  if (do_relu) {
#pragma unroll
    for (int r = 0; r < 8; ++r) c[r] = fmaxf(c[r], 0.f);
  }
#pragma unroll
  for (int r = 0; r < 8; ++r)
    C[(size_t)(row0 + r + 8 * half) * ncols + col0 + l] = c[r];
}

// ---------------------------------------------------------------------------
// Zero agg / attn_sum, reset global max.
// ---------------------------------------------------------------------------
__global__ void init_ws_kernel(float* __restrict__ agg,
                               float* __restrict__ attn_sum,
                               unsigned* __restrict__ gmax, int n) {
  int t = blockIdx.x * blockDim.x + threadIdx.x;
  int nagg = n * 128;
  if (t < nagg) agg[t] = 0.f;
  else if (t < nagg + n * 4) attn_sum[t - nagg] = 0.f;
  if (t == 0) *gmax = 0u;                    // 0u < fenc(anything real)
}

// ---------------------------------------------------------------------------
// Pass 1: one thread per (edge, head): attn = leaky(QK/sqrt(hd) + e@We),
//         wave-reduce max, atomicMax on encoded uint.
// ---------------------------------------------------------------------------
__global__ void edge_attn_kernel(const float* __restrict__ Q,
                                 const float* __restrict__ K,
                                 const int* __restrict__ srcv,
                                 const int* __restrict__ dstv,
                                 const float* __restrict__ edge_attr,
                                 const float* __restrict__ We,
                                 float* __restrict__ attn,
                                 unsigned* __restrict__ gmax, int E) {
  int t = blockIdx.x * blockDim.x + threadIdx.x;
  float a = -3.0e38f;
  if (t < E * 4) {
    int e = t >> 2, h = t & 3;
    int s = srcv[e], d = dstv[e];
    const float4* q = (const float4*)(Q + (size_t)d * 128 + h * 32);
    const float4* k = (const float4*)(K + (size_t)s * 128 + h * 32);
    float acc = 0.f;
#pragma unroll
    for (int i = 0; i < 8; ++i) {
      float4 qa = q[i], ka = k[i];
      acc += qa.x * ka.x + qa.y * ka.y + qa.z * ka.z + qa.w * ka.w;
    }
    acc = acc * SCALE
        + edge_attr[e * 3 + 0] * We[0 * 4 + h]
        + edge_attr[e * 3 + 1] * We[1 * 4 + h]
        + edge_attr[e * 3 + 2] * We[2 * 4 + h];
    a = (acc >= 0.f) ? acc : LEAKY * acc;
    attn[t] = a;
  }
#pragma unroll
  for (int off = 16; off; off >>= 1)         // wave32 max reduction
    a = fmaxf(a, __shfl_xor(a, off, 32));
  if ((threadIdx.x & 31) == 0) atomicMax(gmax, fenc(a));
}

// ---------------------------------------------------------------------------
// Pass 2: attn_exp = exp(attn - gmax); segment_sum into attn_sum[dst,h].
// ---------------------------------------------------------------------------
__global__ void edge_exp_kernel(const int* __restrict__ dstv,
                                float* __restrict__ attn,
                                float* __restrict__ attn_sum,
                                const unsigned* __restrict__ gmax, int E) {
  int t = blockIdx.x * blockDim.x + threadIdx.x;
  if (t >= E * 4) return;
  int e = t >> 2, h = t & 3;
  float m = fdec(*gmax);
  float v = expf(attn[t] - m);
  attn[t] = v;                               // now holds attn_exp
  atomicAdd(&attn_sum[(size_t)dstv[e] * 4 + h], v);
}

// ---------------------------------------------------------------------------
// Pass 3: 128 threads per edge: agg[dst,:] += V[src,:] * attn_norm[h].
// Coalesced V gather + coalesced fp32 atomics (stay in L2: agg is 25.6 MB).
// ---------------------------------------------------------------------------
__global__ void edge_scatter_kernel(const float* __restrict__ V,
                                    const int* __restrict__ srcv,
                                    const int* __restrict__ dstv,
                                    const float* __restrict__ attn,
                                    const float* __restrict__ attn_sum,
                                    float* __restrict__ agg, int E) {
  long long t = (long long)blockIdx.x * blockDim.x + threadIdx.x;
  int e = (int)(t >> 7);
  if (e >= E) return;
  int d = (int)(t & 127);
  int h = d >> 5;
  int s = srcv[e], dd = dstv[e];
  float w = attn[e * 4 + h] / fmaxf(attn_sum[(size_t)dd * 4 + h], 1e-12f);
  atomicAdd(&agg[(size_t)dd * 128 + d], V[(size_t)s * 128 + d] * w);
}

// ---------------------------------------------------------------------------
extern "C" void kernel_launch(void* const* d_in, const int* in_sizes, int n_in,
                              void* d_out, int out_size, void* d_ws, size_t ws_size,
                              hipStream_t stream) {
  const float* x         = (const float*)d_in[0];
  const int*   edge_idx  = (const int*)d_in[1];
  const float* edge_attr = (const float*)d_in[2];
  const float* Wq        = (const float*)d_in[3];
  const float* Wk        = (const float*)d_in[4];
  const float* Wv        = (const float*)d_in[5];
  const float* We        = (const float*)d_in[6];
  const float* Wo        = (const float*)d_in[7];
  const float* bo        = (const float*)d_in[8];
  const float* Wm        = (const float*)d_in[9];
  const float* bm        = (const float*)d_in[10];
  float* out = (float*)d_out;

  const int n = in_sizes[0] / 128;           // 50000
  const int E = in_sizes[2] / 3;             // 800000
  const int* srcv = edge_idx;                // edge_index[0, :]
  const int* dstv = edge_idx + E;            // edge_index[1, :]

  // workspace layout (floats)
  float* ws       = (float*)d_ws;
  float* Q        = ws;
  float* K        = Q        + (size_t)n * 128;
  float* V        = K        + (size_t)n * 128;
  float* agg      = V        + (size_t)n * 128;
  float* agg2     = agg      + (size_t)n * 128;
  float* attn_sum = agg2     + (size_t)n * 128;
  float* attn     = attn_sum + (size_t)n * 4;
  unsigned* gmax  = (unsigned*)(attn + (size_t)E * 4);

  // 0) init accumulators
  {
    int tot = n * 128 + n * 4;
    init_ws_kernel<<<(tot + 255) / 256, 256, 0, stream>>>(agg, attn_sum, gmax, n);
  }

  // 1) Q/K/V = x @ W{q,k,v}   (one wave per 16x16 tile, 8 waves/block)
  const int rtiles = (n + 15) / 16;
  const int gemm_blocks = (rtiles * 8 * 32 + 255) / 256;   // ncols=128 -> 8 ctiles
  wmma_gemm_f32<<<gemm_blocks, 256, 0, stream>>>(x, Wq, nullptr, nullptr, nullptr, Q, n, 128, 128, 0);
  wmma_gemm_f32<<<gemm_blocks, 256, 0, stream>>>(x, Wk, nullptr, nullptr, nullptr, K, n, 128, 128, 0);
  wmma_gemm_f32<<<gemm_blocks, 256, 0, stream>>>(x, Wv, nullptr, nullptr, nullptr, V, n, 128, 128, 0);

  // 2) attention logits + global max
  edge_attn_kernel<<<(E * 4 + 255) / 256, 256, 0, stream>>>(
      Q, K, srcv, dstv, edge_attr, We, attn, gmax, E);

  // 3) exp + segment softmax denominator
  edge_exp_kernel<<<(E * 4 + 255) / 256, 256, 0, stream>>>(dstv, attn, attn_sum, gmax, E);

  // 4) weighted-V scatter into agg
  {
    long long tot = (long long)E * 128;
    edge_scatter_kernel<<<(unsigned)((tot + 255) / 256), 256, 0, stream>>>(
        V, srcv, dstv, attn, attn_sum, agg, E);
  }

  // 5) agg2 = agg @ Wo + bo
  wmma_gemm_f32<<<gemm_blocks, 256, 0, stream>>>(agg, Wo, nullptr, nullptr, bo, agg2, n, 128, 128, 0);

  // 6) out = relu(x @ Wm[0:128] + agg2 @ Wm[128:256] + bm)
  wmma_gemm_f32<<<gemm_blocks, 256, 0, stream>>>(x, Wm, agg2, Wm + (size_t)128 * 128,
                                                 bm, out, n, 128, 128, 1);
}